// MixtralSparseMoeBlock_2723009266293
// MI455X (gfx1250) — compile-verified
//
#include <hip/hip_runtime.h>
#include <cstdint>
#include <cstddef>

// ---------------- problem constants (from reference) ----------------
constexpr int Td = 8192;   // B*S tokens
constexpr int Hd = 2048;   // HIDDEN
constexpr int Fd = 4096;   // FFN
constexpr int Ed = 8;      // experts
// padded compact row space: sum_e align128(N_e) <= 16384 + 8*127 -> 17408
constexpr int ROWCAP    = 17408;          // 136 * 128
constexpr int ROW_TILES = ROWCAP / 128;   // 136

// GEMM tiling: tall tiles so each weight byte is re-read from L2 only
// ~16x per expert instead of ~128x (weights dominate traffic).
constexpr int BM = 128;   // rows per block
constexpr int BN = 64;    // weight rows (output cols) per block
constexpr int BK = 32;    // K chunk staged in LDS

typedef float v2f __attribute__((ext_vector_type(2)));
typedef float v8f __attribute__((ext_vector_type(8)));
typedef int   v4i __attribute__((ext_vector_type(4)));

// ---------------- async global->LDS path (CDNA5) ----------------
#if defined(__has_builtin)
#if __has_builtin(__builtin_amdgcn_global_load_async_to_lds_b128) && \
    __has_builtin(__builtin_amdgcn_s_wait_asynccnt)
#define USE_ASYNC 1
#endif
#endif
#ifndef USE_ASYNC
#define USE_ASYNC 0
#endif

__device__ __forceinline__ void cp16(float* dst_lds, const float* src_g) {
#if USE_ASYNC
    __builtin_amdgcn_global_load_async_to_lds_b128(
        (__attribute__((address_space(1))) v4i*)src_g,
        (__attribute__((address_space(3))) v4i*)dst_lds,
        0, 0);
#else
    *(float4*)dst_lds = *(const float4*)src_g;
#endif
}

__device__ __forceinline__ void tile_commit() {
#if USE_ASYNC
    __builtin_amdgcn_s_wait_asynccnt(0);   // own wave's async copies done
#endif
    __syncthreads();                       // all waves' copies visible
}

// =====================================================================
// 1. Router: logits = x @ gate_w.T ; softmax ; top-2 ; renormalize.
// =====================================================================
__global__ __launch_bounds__(256) void moe_router_kernel(
    const float* __restrict__ x, const float* __restrict__ gate_w,
    int* __restrict__ cnt, int* __restrict__ eidx, float* __restrict__ ew)
{
    __shared__ float g[Ed * Hd];   // 64KB
    const int tid = threadIdx.x;

    const float4* gsrc = (const float4*)gate_w;
    float4* gdst = (float4*)g;
    #pragma unroll
    for (int i = 0; i < 16; ++i) gdst[tid + i * 256] = gsrc[tid + i * 256];
    __syncthreads();

    const int t = blockIdx.x * 256 + tid;
    float acc[Ed];
    #pragma unroll
    for (int e = 0; e < Ed; ++e) acc[e] = 0.f;

    const float* xr = x + (size_t)t * Hd;
    for (int h = 0; h < Hd; h += 4) {
        const float4 xv = *(const float4*)(xr + h);
        #pragma unroll
        for (int e = 0; e < Ed; ++e) {
            const float4 gv = *(const float4*)&g[e * Hd + h];
            acc[e] += xv.x * gv.x + xv.y * gv.y + xv.z * gv.z + xv.w * gv.w;
        }
    }

    float l1 = -3.4e38f, l2 = -3.4e38f;
    int i1 = 0, i2 = 0;
    #pragma unroll
    for (int e = 0; e < Ed; ++e) {
        float v = acc[e];
        if (v > l1)      { l2 = l1; i2 = i1; l1 = v; i1 = e; }
        else if (v > l2) { l2 = v;  i2 = e; }
    }
    const float d  = __expf(l2 - l1);
    const float p1 = 1.0f / (1.0f + d);
    const float p2 = d * p1;

    eidx[2 * t]     = i1;  ew[2 * t]     = p1;
    eidx[2 * t + 1] = i2;  ew[2 * t + 1] = p2;
    atomicAdd(&cnt[i1], 1);
    atomicAdd(&cnt[i2], 1);
}

// =====================================================================
// 2. Scan: 128-aligned exclusive scan of counts.
// =====================================================================
__global__ void moe_scan_kernel(const int* __restrict__ cnt,
                                int* __restrict__ cursor,
                                int* __restrict__ offs)
{
    if (threadIdx.x == 0 && blockIdx.x == 0) {
        int off = 0;
        for (int e = 0; e < Ed; ++e) {
            offs[e] = off;
            off += (cnt[e] + 127) & ~127;
            cursor[e] = 0;
        }
        offs[Ed] = off;
    }
}

// =====================================================================
// 3. Assign tokens into padded per-expert row space.
// =====================================================================
__global__ __launch_bounds__(256) void moe_assign_kernel(
    const int* __restrict__ eidx, const float* __restrict__ ew,
    const int* __restrict__ offs, int* __restrict__ cursor,
    int* __restrict__ rtok, float* __restrict__ rw)
{
    const int t = blockIdx.x * 256 + threadIdx.x;
    #pragma unroll
    for (int k = 0; k < 2; ++k) {
        const int e = eidx[2 * t + k];
        const int slot = atomicAdd(&cursor[e], 1);
        const int row = offs[e] + slot;
        rtok[row] = t;
        rw[row]   = ew[2 * t + k];
    }
}

__device__ __forceinline__ int find_expert(const int* offs, int rowbase) {
    int e = 0;
    #pragma unroll
    for (int i = 1; i < Ed; ++i)
        if (rowbase >= offs[i]) e = i;
    return e;
}

// =====================================================================
// 4. FFN1: act = relu(X @ w1[e]^T) * (X @ w3[e]^T)
//    128x64 block tile, 8 waves x (32x32) macro-tiles, fp32 WMMA.
// =====================================================================
__global__ __launch_bounds__(256) void moe_ffn1_kernel(
    const float* __restrict__ x, const float* __restrict__ w1,
    const float* __restrict__ w3,
    const int* __restrict__ offs, const int* __restrict__ cnt,
    const int* __restrict__ rtok, float* __restrict__ act)
{
    __shared__ float As[BM * BK];    // 16KB
    __shared__ float B1s[BN * BK];   // 8KB
    __shared__ float B3s[BN * BK];   // 8KB
    __shared__ int   stok[BM];

    const int tid = threadIdx.x;
    const int rowbase = blockIdx.x * BM;
    if (rowbase >= offs[Ed]) return;                 // uniform
    const int e = find_expert(offs, rowbase);
    const int ne = cnt[e];
    const int local0 = rowbase - offs[e];
    if (local0 >= ne) return;                        // uniform

    // token ids; invalid rows alias token 0 (their outputs are masked)
    if (tid < BM) stok[tid] = (local0 + tid < ne) ? rtok[rowbase + tid] : 0;
    __syncthreads();

    const int n0   = blockIdx.y * BN;
    const int wid  = tid >> 5;
    const int lane = tid & 31;
    const int half = lane >> 4;
    const int lm   = lane & 15;
    const int wm   = (wid & 3) * 32;   // wave M base (4 groups of 32)
    const int wn   = (wid >> 2) * 32;  // wave N base (2 groups of 32)

    // A loaders: 2 threads/row, 16 floats each
    const int arow = tid >> 1;
    const int akb  = (tid & 1) * 16;
    const float* asrc = x + (size_t)stok[arow] * Hd + akb;
    // B loaders: 4 threads/row, 8 floats each
    const int brow = tid >> 2;
    const int bkb  = (tid & 3) * 8;
    const size_t wrow = ((size_t)e * Fd + n0 + brow) * Hd + bkb;

    v8f a1[2][2] = {}, a3[2][2] = {};

    for (int h0 = 0; h0 < Hd; h0 += BK) {
        #pragma unroll
        for (int j = 0; j < 4; ++j)
            cp16(&As[arow * BK + akb + j * 4], asrc + h0 + j * 4);
        #pragma unroll
        for (int j = 0; j < 2; ++j) {
            cp16(&B1s[brow * BK + bkb + j * 4], w1 + wrow + h0 + j * 4);
            cp16(&B3s[brow * BK + bkb + j * 4], w3 + wrow + h0 + j * 4);
        }
        __builtin_prefetch(w1 + wrow + h0 + BK, 0, 3);
        __builtin_prefetch(w3 + wrow + h0 + BK, 0, 3);
        tile_commit();

        #pragma unroll
        for (int k4 = 0; k4 < BK / 4; ++k4) {
            const int ko = k4 * 4 + half * 2;
            v2f af[2], b1f[2], b3f[2];
            #pragma unroll
            for (int mi = 0; mi < 2; ++mi)
                af[mi] = *(const v2f*)&As[(wm + mi * 16 + lm) * BK + ko];
            #pragma unroll
            for (int nj = 0; nj < 2; ++nj) {
                b1f[nj] = *(const v2f*)&B1s[(wn + nj * 16 + lm) * BK + ko];
                b3f[nj] = *(const v2f*)&B3s[(wn + nj * 16 + lm) * BK + ko];
            }
            #pragma unroll
            for (int mi = 0; mi < 2; ++mi)
                #pragma unroll
                for (int nj = 0; nj < 2; ++nj) {
                    a1[mi][nj] = __builtin_amdgcn_wmma_f32_16x16x4_f32(
                        false, af[mi], false, b1f[nj], (short)0, a1[mi][nj],
                        false, false);
                    a3[mi][nj] = __builtin_amdgcn_wmma_f32_16x16x4_f32(
                        false, af[mi], false, b3f[nj], (short)0, a3[mi][nj],
                        false, false);
                }
        }
        __syncthreads();
    }

    // D layout: VGPR i, lanes 0-15 -> M=i, lanes 16-31 -> M=8+i; N = lm
    #pragma unroll
    for (int mi = 0; mi < 2; ++mi)
        #pragma unroll
        for (int nj = 0; nj < 2; ++nj)
            #pragma unroll
            for (int i = 0; i < 8; ++i) {
                const int m = wm + mi * 16 + i + half * 8;
                if (local0 + m < ne) {
                    float g1 = a1[mi][nj][i];
                    g1 = g1 > 0.f ? g1 : 0.f;
                    act[(size_t)(rowbase + m) * Fd + n0 + wn + nj * 16 + lm] =
                        g1 * a3[mi][nj][i];
                }
            }
}

// =====================================================================
// 5. FFN2: out[token] += route_w * (act @ w2[e]^T)
// =====================================================================
__global__ __launch_bounds__(256) void moe_ffn2_kernel(
    const float* __restrict__ act, const float* __restrict__ w2,
    const int* __restrict__ offs, const int* __restrict__ cnt,
    const int* __restrict__ rtok, const float* __restrict__ rw,
    float* __restrict__ out)
{
    __shared__ float As[BM * BK];
    __shared__ float Bs[BN * BK];
    __shared__ int   stok[BM];
    __shared__ float sw[BM];

    const int tid = threadIdx.x;
    const int rowbase = blockIdx.x * BM;
    if (rowbase >= offs[Ed]) return;
    const int e = find_expert(offs, rowbase);
    const int ne = cnt[e];
    const int local0 = rowbase - offs[e];
    if (local0 >= ne) return;

    if (tid < BM) {
        const bool v = (local0 + tid) < ne;
        stok[tid] = v ? rtok[rowbase + tid] : 0;
        sw[tid]   = v ? rw[rowbase + tid]   : 0.f;
    }
    __syncthreads();

    const int n0   = blockIdx.y * BN;
    const int wid  = tid >> 5;
    const int lane = tid & 31;
    const int half = lane >> 4;
    const int lm   = lane & 15;
    const int wm   = (wid & 3) * 32;
    const int wn   = (wid >> 2) * 32;

    const int arow = tid >> 1;
    const int akb  = (tid & 1) * 16;
    const float* asrc = act + (size_t)(rowbase + arow) * Fd + akb;
    const int brow = tid >> 2;
    const int bkb  = (tid & 3) * 8;
    const size_t wrow = ((size_t)e * Hd + n0 + brow) * Fd + bkb;

    v8f acc[2][2] = {};

    for (int f0 = 0; f0 < Fd; f0 += BK) {
        #pragma unroll
        for (int j = 0; j < 4; ++j)
            cp16(&As[arow * BK + akb + j * 4], asrc + f0 + j * 4);
        #pragma unroll
        for (int j = 0; j < 2; ++j)
            cp16(&Bs[brow * BK + bkb + j * 4], w2 + wrow + f0 + j * 4);
        __builtin_prefetch(w2 + wrow + f0 + BK, 0, 3);
        tile_commit();

        #pragma unroll
        for (int k4 = 0; k4 < BK / 4; ++k4) {
            const int ko = k4 * 4 + half * 2;
            v2f af[2], bf[2];
            #pragma unroll
            for (int mi = 0; mi < 2; ++mi)
                af[mi] = *(const v2f*)&As[(wm + mi * 16 + lm) * BK + ko];
            #pragma unroll
            for (int nj = 0; nj < 2; ++nj)
                bf[nj] = *(const v2f*)&Bs[(wn + nj * 16 + lm) * BK + ko];
            #pragma unroll
            for (int mi = 0; mi < 2; ++mi)
                #pragma unroll
                for (int nj = 0; nj < 2; ++nj)
                    acc[mi][nj] = __builtin_amdgcn_wmma_f32_16x16x4_f32(
                        false, af[mi], false, bf[nj], (short)0, acc[mi][nj],
                        false, false);
        }
        __syncthreads();
    }

    #pragma unroll
    for (int mi = 0; mi < 2; ++mi)
        #pragma unroll
        for (int nj = 0; nj < 2; ++nj)
            #pragma unroll
            for (int i = 0; i < 8; ++i) {
                const int m = wm + mi * 16 + i + half * 8;
                if (local0 + m < ne) {
                    const int   t = stok[m];
                    const float w = sw[m];
                    atomicAdd(&out[(size_t)t * Hd + n0 + wn + nj * 16 + lm],
                              acc[mi][nj][i] * w);
                }
            }
}

// =====================================================================
// launch
// =====================================================================
extern "C" void kernel_launch(void* const* d_in, const int* in_sizes, int n_in,
                              void* d_out, int out_size, void* d_ws, size_t ws_size,
                              hipStream_t stream) {
    (void)in_sizes; (void)n_in; (void)out_size; (void)ws_size;

    const float* x      = (const float*)d_in[0];   // [T, H]
    const float* gate_w = (const float*)d_in[1];   // [E, H]
    const float* w1     = (const float*)d_in[2];   // [E, F, H]
    const float* w2     = (const float*)d_in[3];   // [E, H, F]
    const float* w3     = (const float*)d_in[4];   // [E, F, H]
    float* out = (float*)d_out;                    // [T, H]

    // ---- workspace layout ----
    int*   cnt    = (int*)d_ws;                     // [8]
    int*   cursor = cnt + 8;                        // [8]
    int*   offs   = cnt + 16;                       // [9] (pad to 16)
    int*   eidx   = cnt + 32;                       // [2T]
    float* ew     = (float*)(eidx + 2 * Td);        // [2T]
    int*   rtok   = (int*)(ew + 2 * Td);            // [ROWCAP]
    float* rw     = (float*)(rtok + ROWCAP);        // [ROWCAP]
    uintptr_t actp = (uintptr_t)(rw + ROWCAP);
    actp = (actp + 255) & ~(uintptr_t)255;
    float* act = (float*)actp;                      // [ROWCAP, Fd] fp32

    (void)hipMemsetAsync(d_ws, 0, 128, stream);
    (void)hipMemsetAsync(out, 0, (size_t)Td * Hd * sizeof(float), stream);

    moe_router_kernel<<<Td / 256, 256, 0, stream>>>(x, gate_w, cnt, eidx, ew);
    moe_scan_kernel<<<1, 32, 0, stream>>>(cnt, cursor, offs);
    moe_assign_kernel<<<Td / 256, 256, 0, stream>>>(eidx, ew, offs, cursor, rtok, rw);

    moe_ffn1_kernel<<<dim3(ROW_TILES, Fd / BN), 256, 0, stream>>>(
        x, w1, w3, offs, cnt, rtok, act);
    moe_ffn2_kernel<<<dim3(ROW_TILES, Hd / BN), 256, 0, stream>>>(
        act, w2, offs, cnt, rtok, rw, out);
}